// LSTM_net_66005057405422
// MI455X (gfx1250) — compile-verified
//
#include <hip/hip_runtime.h>
#include <hip/hip_bf16.h>

// ---------------------------------------------------------------------------
// 50-layer bidirectional LSTM (H=5, proj=1, G=20, B=16, T=8192) for gfx1250.
//
// Latency-bound problem: 409,600 serially dependent cell steps. Each step is
// vectorized over the 16-batch dimension with V_WMMA_F32_16X16X4_F32:
//   A (16x4)  = per-batch row [in0, in1, h, 1]
//   B (4x16)  = per-gate col  [Wih0, Wih1, Whh, bih+bhh]
//   D (16x16) = gate pre-activations, lane = gate, VGPR = batch
// Two WMMAs per step cover gate tiles [i0..i4,f0..f4,g0..g4,pad] and [o0..o4].
// Nonlinearities: native v_tanh_f32 (sigmoid(x) = 0.5*tanh(0.5x)+0.5).
// Cross-lane traffic: ds_bpermute for f/g gathers, 3-round xor butterfly for
// the 5-wide Whr projection, and the h broadcast into next step's A matrix.
// WGP-scope global_prefetch_b8 (inline asm, scope 0 = fill all cache levels,
// ISA §10.5) pulls upcoming activations from GL2 into L0 ahead of the chain.
// ---------------------------------------------------------------------------

#define TT 8192
#define BB 16
#define LL 50
#define HH 5
#define GG 20

typedef float v2f __attribute__((ext_vector_type(2)));
typedef float v8f __attribute__((ext_vector_type(8)));

#if __has_builtin(__builtin_amdgcn_tanhf)
__device__ __forceinline__ float fast_tanh(float x) { return __builtin_amdgcn_tanhf(x); }
#elif __has_builtin(__builtin_amdgcn_tanh_f32)
__device__ __forceinline__ float fast_tanh(float x) { return __builtin_amdgcn_tanh_f32(x); }
#else
__device__ __forceinline__ float fast_tanh(float x) {
    float r;
    asm("v_tanh_f32 %0, %1" : "=v"(r) : "v"(x));
    return r;
}
#endif

__device__ __forceinline__ float bperm(int srcLane, float v) {
    return __int_as_float(__builtin_amdgcn_ds_bpermute(srcLane << 2, __float_as_int(v)));
}

// WGP-scope prefetch: scope field 0 (no modifier) pulls into all cache levels
// on miss (ISA §10.5); __builtin_prefetch can only reach SE/SYS scopes, which
// stop at GL2 and are useless for L0-latency hiding.
__device__ __forceinline__ void prefetch_wgp(const void* p) {
    asm volatile("global_prefetch_b8 %0, off" :: "v"(p));
}

// Stage x [B,T] into layer-io layout [T][B][2] with in1 = 0 (layer-0 input).
__global__ void lstm_stage_input(const float* __restrict__ x, float* __restrict__ buf0) {
    int i = blockIdx.x * blockDim.x + threadIdx.x;
    if (i >= BB * TT) return;
    int b = i / TT, t = i % TT;
    buf0[(t * BB + b) * 2 + 0] = x[b * TT + t];
    buf0[(t * BB + b) * 2 + 1] = 0.0f;
}

// Persistent scan: 1 block, 2 waves (wave 0 = fwd, wave 1 = bwd), all layers.
__global__ void __launch_bounds__(64, 1)
lstm_scan(const float* __restrict__ Wih0, const float* __restrict__ WihR,
          const float* __restrict__ Whh,  const float* __restrict__ bih,
          const float* __restrict__ bhh,  const float* __restrict__ Whr,
          float* __restrict__ buf0, float* __restrict__ buf1) {
    const int lane = threadIdx.x & 31;
    const int dir  = threadIdx.x >> 5;     // wave 0: forward, wave 1: backward
    const int n    = lane & 15;            // WMMA column (gate) / batch index
    const int half = lane >> 4;            // A/B K-half: 0 -> K0/K1, 1 -> K2/K3

    // tile0 col n -> gate n (n<15: i0..i4,f0..f4,g0..g4), col15 = zero pad
    // tile1 col n -> gate 15+n (n<5: o0..o4), cols 5..15 = zero pad
    const int   g0 = (n < 15) ? n : 0;
    const float v0 = (n < 15) ? 1.0f : 0.0f;
    const int   g1 = (n < 5) ? (15 + n) : 0;
    const float v1 = (n < 5) ? 1.0f : 0.0f;

    // tile0 activation constants: sigmoid for i/f (cols 0..9), tanh for g (10..14)
    const bool  tgate = (n >= 10 && n <= 14);
    const float sc0 = tgate ? 1.0f : 0.5f;     // also the output scale m (m == sc)
    const float q0  = tgate ? 0.0f : 0.5f;

    for (int l = 0; l < LL; ++l) {
        const int ld = l * 2 + dir;

        // --- Build B matrices for this (layer, dir). Assumed B VGPR layout
        // mirrors the documented A 16x4 layout: VGPR0 = {K0 | K2}, VGPR1 = {K1 | K3}.
        auto loadB = [&](int g, float valid) -> v2f {
            float wih0, wih1;
            if (l == 0) { wih0 = Wih0[dir * GG + g]; wih1 = 0.0f; }
            else        { const float* p = WihR + (((l - 1) * 2 + dir) * GG + g) * 2;
                          wih0 = p[0]; wih1 = p[1]; }
            float whh = Whh[ld * GG + g];
            float bs  = bih[ld * GG + g] + bhh[ld * GG + g];
            v2f bm;
            bm.x = (half ? whh : wih0) * valid;  // rows K0 / K2
            bm.y = (half ? bs  : wih1) * valid;  // rows K1 / K3
            return bm;
        };
        const v2f B0 = loadB(g0, v0);
        const v2f B1 = loadB(g1, v1);
        const float whr = (n < 5) ? Whr[ld * HH + n] : 0.0f;  // zero-pads reduction

        const float* rb = (l & 1) ? buf1 : buf0;
        float*       wb = (l & 1) ? buf0 : buf1;
        const float2* rin = reinterpret_cast<const float2*>(rb);

        float hAll = 0.0f;      // every lane: h[lane&15]
        float c[8];             // lane k (<5) of each half: c[batch=r+8*half][k]
#pragma unroll
        for (int r = 0; r < 8; ++r) c[r] = 0.0f;

        for (int s = 0; s < TT; ++s) {
            const int t = dir ? (TT - 1 - s) : s;

            // Layer input for this timestep (GL2-resident). Prefetch 8 steps
            // ahead at WGP scope so the 128B line is in L0 before the
            // dependent chain reaches it. tp is clamped (not branched) so the
            // inner loop stays branch-free with EXEC untouched.
            const float2 xin = rin[t * BB + n];
            {
                int tp = dir ? (t - 8) : (t + 8);
                tp = tp < 0 ? 0 : (tp >= TT ? TT - 1 : tp);
                prefetch_wgp(&rin[tp * BB + n]);
            }

            // A matrix: lanes 0-15 = {in0,in1}, lanes 16-31 = {h, 1}.
            v2f a;
            a.x = half ? hAll : xin.x;
            a.y = half ? 1.0f : xin.y;

            const v8f cz = {};
            v8f d0 = __builtin_amdgcn_wmma_f32_16x16x4_f32(
                false, a, false, B0, (short)0, cz, false, false);
            v8f d1 = __builtin_amdgcn_wmma_f32_16x16x4_f32(
                false, a, false, B1, (short)0, cz, false, false);

            // Activations: act = m*tanh(m*x)+q with m==sc (sigmoid or tanh per lane).
            float act0[8], act1[8];
#pragma unroll
            for (int r = 0; r < 8; ++r) {
                act0[r] = fmaf(sc0,  fast_tanh(sc0 * d0[r]),  q0);
                act1[r] = fmaf(0.5f, fast_tanh(0.5f * d1[r]), 0.5f);
            }

            // Cell update + projection contribution. Lane k (<5) holds cell k;
            // f comes from lane k+5, g from lane k+10 (same half). Junk lanes
            // stay bounded and are zeroed by whr==0 before the reduction.
            float p[8];
#pragma unroll
            for (int r = 0; r < 8; ++r) {
                const float fv = bperm(lane + 5,  act0[r]);
                const float gv = bperm(lane + 10, act0[r]);
                c[r] = fmaf(fv, c[r], act0[r] * gv);
                p[r] = act1[r] * fast_tanh(c[r]) * whr;
            }

            // h[batch] = sum_k p[k]: 3-round xor butterfly over the 8-lane group
            // (lanes 5..7 carry exact zeros). Result lands in lanes 0-7 / 16-23.
#pragma unroll
            for (int m = 1; m <= 4; m <<= 1) {
#pragma unroll
                for (int r = 0; r < 8; ++r) p[r] += bperm(lane ^ m, p[r]);
            }

            // Gather: every lane picks h[lane&15] (reg r = b&7, src half = b>>3).
            float hN = 0.0f;
            const int srcl = (lane & 8) ? 16 : 0;
#pragma unroll
            for (int r = 0; r < 8; ++r) {
                const float bp = bperm(srcl, p[r]);
                hN = ((lane & 7) == r) ? bp : hN;
            }
            hAll = hN;

            if (lane < 16)
                wb[(t * BB + lane) * 2 + dir] = hAll;
        }

        __threadfence();
        __syncthreads();
    }
}

// softmax over (ss, 1-ss): p0 = sigmoid(2*ss - 1), p1 = 1 - p0. Output [B,2,T].
__global__ void lstm_epilogue(const float* __restrict__ fin, float* __restrict__ out) {
    int i = blockIdx.x * blockDim.x + threadIdx.x;
    if (i >= BB * TT) return;
    int b = i / TT, t = i % TT;
    float ss = fin[(t * BB + b) * 2 + 0] + fin[(t * BB + b) * 2 + 1];
    float p0 = 1.0f / (1.0f + __expf(1.0f - 2.0f * ss));
    out[b * 2 * TT + t]      = p0;
    out[b * 2 * TT + TT + t] = 1.0f - p0;
}

extern "C" void kernel_launch(void* const* d_in, const int* in_sizes, int n_in,
                              void* d_out, int out_size, void* d_ws, size_t ws_size,
                              hipStream_t stream) {
    const float* x     = (const float*)d_in[0];   // [16, 8192]
    const float* Wih0  = (const float*)d_in[1];   // [2, 20, 1]
    const float* WihR  = (const float*)d_in[2];   // [49, 2, 20, 2]
    const float* Whh   = (const float*)d_in[3];   // [50, 2, 20, 1]
    const float* bih   = (const float*)d_in[4];   // [50, 2, 20]
    const float* bhh   = (const float*)d_in[5];   // [50, 2, 20]
    const float* Whr   = (const float*)d_in[6];   // [50, 2, 1, 5]
    float* out = (float*)d_out;                   // [16, 2, 8192]

    float* buf0 = (float*)d_ws;                   // [T][B][2] ping
    float* buf1 = buf0 + (size_t)TT * BB * 2;     // [T][B][2] pong (2 MB total)

    const int nTB = BB * TT;
    lstm_stage_input<<<(nTB + 255) / 256, 256, 0, stream>>>(x, buf0);
    lstm_scan<<<1, 64, 0, stream>>>(Wih0, WihR, Whh, bih, bhh, Whr, buf0, buf1);
    // 50 layers: final result lands back in buf0 (even layer count).
    lstm_epilogue<<<(nTB + 255) / 256, 256, 0, stream>>>(buf0, out);
}